// NonMaximumSuppression_38843684225723
// MI455X (gfx1250) — compile-verified
//
#include <hip/hip_runtime.h>
#include <hip/hip_bf16.h>
#include <stdint.h>

#define N_ANCH      262144
#define NUM_CLASSES 80
#define DET_DIM     84
#define TOPK        1000
#define KPAD        1024
#define MAX_BOXES   300
#define NMS_THR     0.4f
#define CAND_CAP    2048

typedef __attribute__((ext_vector_type(2))) float v2f;
typedef __attribute__((ext_vector_type(8))) float v8f;

// ---------------- zero scratch (hist1 | hist2 | stage, contiguous) ----------
__global__ void k_zero(uint32_t* __restrict__ p, int n) {
    int i = blockIdx.x * blockDim.x + threadIdx.x;
    if (i < n) p[i] = 0u;
}

// ---------------- pass 1: class-max -> sortable u32 + 16-bit histogram ------
__global__ void k_scores(const float* __restrict__ cls,
                         uint32_t* __restrict__ su,
                         uint32_t* __restrict__ hist1) {
    int i = blockIdx.x * blockDim.x + threadIdx.x;
    if (i >= N_ANCH) return;
    if (i + 2048 < N_ANCH)
        __builtin_prefetch(cls + (size_t)(i + 2048) * NUM_CLASSES, 0, 0);
    const float4* row = reinterpret_cast<const float4*>(cls + (size_t)i * NUM_CLASSES);
    float m = -__builtin_inff();
#pragma unroll
    for (int q = 0; q < NUM_CLASSES / 4; ++q) {
        float4 v = row[q];
        m = fmaxf(m, fmaxf(fmaxf(v.x, v.y), fmaxf(v.z, v.w)));
    }
    uint32_t b = __float_as_uint(m);
    uint32_t u = (b & 0x80000000u) ? ~b : (b | 0x80000000u); // order-preserving
    su[i] = u;
    atomicAdd(&hist1[u >> 16], 1u);
}

// ---------------- radix-select step: find crossing bucket from the top ------
__global__ __launch_bounds__(1024)
void k_findbucket(const uint32_t* __restrict__ hist,
                  const uint32_t* __restrict__ basePtr,
                  uint32_t* __restrict__ outB,
                  uint32_t* __restrict__ outAbove) {
    __shared__ uint32_t csum[1024];
    int t = threadIdx.x;
    uint32_t s = 0;
    const uint32_t* h = hist + t * 64;
#pragma unroll 8
    for (int q = 0; q < 64; ++q) s += h[q];
    csum[t] = s;
    __syncthreads();
    if (t == 0) {
        uint32_t base = basePtr ? *basePtr : 0u;
        uint32_t cum = base, B = 0, above = base;
        bool found = false;
        for (int c = 1023; c >= 0 && !found; --c) {
            if (cum + csum[c] >= (uint32_t)TOPK) {
                for (int b2 = c * 64 + 63; b2 >= c * 64; --b2) {
                    uint32_t hv = hist[b2];
                    if (cum + hv >= (uint32_t)TOPK) { B = (uint32_t)b2; above = cum; found = true; break; }
                    cum += hv;
                }
            } else {
                cum += csum[c];
            }
        }
        *outB = B;
        *outAbove = above;
    }
}

// ---------------- pass 2 histogram: low 16 bits within crossing bucket ------
__global__ void k_hist2(const uint32_t* __restrict__ su,
                        const uint32_t* __restrict__ stage,
                        uint32_t* __restrict__ hist2) {
    int i = blockIdx.x * blockDim.x + threadIdx.x;
    if (i >= N_ANCH) return;
    uint32_t u = su[i];
    if ((u >> 16) == stage[0]) atomicAdd(&hist2[u & 0xFFFFu], 1u);
}

// ---------------- collect all candidates >= kth value -----------------------
__global__ void k_collect(const uint32_t* __restrict__ su,
                          const uint32_t* __restrict__ stage,
                          unsigned long long* __restrict__ cand,
                          uint32_t* __restrict__ counter) {
    int i = blockIdx.x * blockDim.x + threadIdx.x;
    if (i >= N_ANCH) return;
    uint32_t u = su[i];
    uint32_t kth = (stage[0] << 16) | stage[2];
    if (u >= kth) {
        uint32_t p = atomicAdd(counter, 1u);
        if (p < CAND_CAP)
            cand[p] = ((unsigned long long)u << 32) | (uint32_t)(~(uint32_t)i);
    }
}

// ---------------- bitonic sort 2048 keys desc, gather top-1000 boxes --------
__global__ __launch_bounds__(1024)
void k_sortgather(const unsigned long long* __restrict__ cand,
                  const uint32_t* __restrict__ stage,
                  const float* __restrict__ boxes,
                  uint32_t* __restrict__ topk_idx,
                  float4* __restrict__ boxes_k,
                  float* __restrict__ area_k) {
    __shared__ unsigned long long key[CAND_CAP];
    int t = threadIdx.x;
    uint32_t cnt = stage[4];
    if (cnt > CAND_CAP) cnt = CAND_CAP;
    for (int p = t; p < CAND_CAP; p += 1024)
        key[p] = (p < (int)cnt) ? cand[p] : 0ull;
    __syncthreads();
    for (unsigned k = 2; k <= CAND_CAP; k <<= 1) {
        for (unsigned j = k >> 1; j > 0; j >>= 1) {
            unsigned blk = (unsigned)t / j, pos = (unsigned)t % j;
            unsigned a = blk * 2u * j + pos, b = a + j;
            bool descBlock = ((a & k) == 0);
            unsigned long long ka = key[a], kb = key[b];
            if ((ka < kb) == descBlock) { key[a] = kb; key[b] = ka; }
            __syncthreads();
        }
    }
    for (int p = t; p < KPAD; p += 1024) {
        if (p < TOPK) {
            unsigned long long kk = key[p];
            uint32_t idx = kk ? (uint32_t)(~(uint32_t)kk) : 0u;
            topk_idx[p] = idx;
            float4 bx = reinterpret_cast<const float4*>(boxes)[idx];
            boxes_k[p] = bx;
            area_k[p] = (bx.z - bx.x) * (bx.w - bx.y);
        } else {
            topk_idx[p] = 0u;
            boxes_k[p] = make_float4(0.f, 0.f, 0.f, 0.f);
            area_k[p] = 0.f;
        }
    }
}

// ---------------- IoU tiles: WMMA outer-sum + ballot-packed bitmask ---------
__global__ __launch_bounds__(32)
void k_iou(const float4* __restrict__ boxes_k,
           const float* __restrict__ area_k,
           uint16_t* __restrict__ overH) {
    int tileC = blockIdx.x, tileR = blockIdx.y;
    int lane = threadIdx.x;              // wave32, EXEC all ones
    int r0 = tileR * 16, c0 = tileC * 16;
    int half = lane >> 4, ln = lane & 15;

    // A (16x4, documented ISA layout): rows M = [area_M, 1, 0, 0] across K.
    //   lanes 0-15:  V0 = K0 = area_M, V1 = K1 = 1
    //   lanes 16-31: V0 = K2 = 0,      V1 = K3 = 0
    // B (4x16): made UNIFORM across half-waves so that under either
    // half-wave<->K convention B's effective rows are [1, area_N, 1, area_N];
    // A's zeros in K2/K3 kill the duplicates:
    //   dot = area_M*1 + 1*area_N + 0 + 0  (exact RNE(area_M + area_N))
    v2f A, B;
    A.x = (half == 0) ? area_k[r0 + ln] : 0.f;
    A.y = (half == 0) ? 1.f : 0.f;
    B.x = 1.f;                 // all lanes: K0 and K2 rows = ones
    B.y = area_k[c0 + ln];     // all lanes: K1 and K3 rows = area_col[N]
    v8f C = {};
    v8f S = __builtin_amdgcn_wmma_f32_16x16x4_f32(false, A, false, B,
                                                  (short)0, C, false, false);

    int jglob = c0 + ln;
    float4 bj = boxes_k[jglob];
#pragma unroll
    for (int r = 0; r < 8; ++r) {
        int iglob = r0 + r + half * 8;          // C/D layout: VGPR r -> M=r / M=r+8
        float4 bi = boxes_k[iglob];
        float ix1 = fmaxf(bi.x, bj.x), iy1 = fmaxf(bi.y, bj.y);
        float ix2 = fminf(bi.z, bj.z), iy2 = fminf(bi.w, bj.w);
        float inter = fmaxf(ix2 - ix1, 0.f) * fmaxf(iy2 - iy1, 0.f);
        float uni = S[r] - inter;
        float iou = inter / fmaxf(uni, 1e-8f);
        bool over = (iou > NMS_THR) && (iglob < TOPK) && (jglob < TOPK);
        uint32_t bal = (uint32_t)__ballot(over); // wave32: low 16 = row r0+r, high 16 = row r0+r+8
        if (lane == 0) {
            overH[(size_t)(r0 + r) * 64 + tileC]     = (uint16_t)(bal & 0xFFFFu);
            overH[(size_t)(r0 + r + 8) * 64 + tileC] = (uint16_t)(bal >> 16);
        }
    }
}

// ---------------- greedy sequential suppression scan: ONE wave32 ------------
__global__ __launch_bounds__(32)
void k_scan(const uint16_t* __restrict__ overH,
            const uint32_t* __restrict__ topk_idx,
            int* __restrict__ kept,
            float* __restrict__ validf) {
    int lane = threadIdx.x;                    // lane l owns columns [32l, 32l+32)
    uint32_t sup = 0u;
    const uint32_t* ovr32 = reinterpret_cast<const uint32_t*>(overH); // row = 32 u32
    for (int i = 0; i < TOPK; ++i) {
        uint32_t w = __shfl(sup, i >> 5, 32);
        bool active = !((w >> (i & 31)) & 1u); // uniform across wave
        if (active) {
            uint32_t row = ovr32[(size_t)i * 32 + lane];
            int lo = i + 1 - 32 * lane;        // mask for j > i
            uint32_t maskGT = (lo <= 0) ? 0xFFFFFFFFu
                            : (lo >= 32) ? 0u : (0xFFFFFFFFu << lo);
            sup |= row & maskGT;
        }
    }
    __shared__ uint32_t ssup[32];
    ssup[lane] = sup;
    __syncthreads();
    if (lane == 0) {
        int cnt = 0;
        int top0 = (int)topk_idx[0];           // reference: order_pad[K]=0 -> d_k[0]*0
        for (int j = 0; j < TOPK && cnt < MAX_BOXES; ++j) {
            if (!((ssup[j >> 5] >> (j & 31)) & 1u)) {
                kept[cnt] = (int)topk_idx[j];
                validf[cnt] = 1.f;
                ++cnt;
            }
        }
        for (int s = cnt; s < MAX_BOXES; ++s) { kept[s] = top0; validf[s] = 0.f; }
    }
}

// ---------------- final gather 300 x 84 -------------------------------------
__global__ void k_out(const float* __restrict__ det,
                      const int* __restrict__ kept,
                      const float* __restrict__ validf,
                      float* __restrict__ out) {
    int x = blockIdx.x * blockDim.x + threadIdx.x;
    if (x >= MAX_BOXES * DET_DIM) return;
    int s = x / DET_DIM, c = x % DET_DIM;
    out[x] = det[(size_t)kept[s] * DET_DIM + c] * validf[s];
}

extern "C" void kernel_launch(void* const* d_in, const int* in_sizes, int n_in,
                              void* d_out, int out_size, void* d_ws, size_t ws_size,
                              hipStream_t stream) {
    const float* boxes = (const float*)d_in[0]; // (1, N, 4)
    const float* cls   = (const float*)d_in[1]; // (1, N, 80)
    const float* det   = (const float*)d_in[2]; // (1, N, 84)
    float* out = (float*)d_out;                 // (1, 300, 84)

    char* ws = (char*)d_ws;
    uint32_t* su              = (uint32_t*)(ws + 0);          // 1 MB
    uint32_t* hist1           = (uint32_t*)(ws + 1048576);    // 256 KB
    uint32_t* hist2           = (uint32_t*)(ws + 1310720);    // 256 KB
    uint32_t* stage           = (uint32_t*)(ws + 1572864);    // 64 B
    unsigned long long* cand  = (unsigned long long*)(ws + 1572928); // 16 KB
    uint32_t* topk            = (uint32_t*)(ws + 1589312);    // 4 KB
    float4*   boxesk          = (float4*)(ws + 1593408);      // 16 KB (16B aligned)
    float*    areak           = (float*)(ws + 1609792);       // 4 KB
    uint16_t* overH           = (uint16_t*)(ws + 1613888);    // 128 KB
    int*      kept            = (int*)(ws + 1744960);         // 1200 B
    float*    validf          = (float*)(ws + 1746160);       // 1200 B

    const int ZN = 65536 * 2 + 16;                            // hist1+hist2+stage (contiguous)
    k_zero<<<(ZN + 255) / 256, 256, 0, stream>>>(hist1, ZN);
    k_scores<<<N_ANCH / 256, 256, 0, stream>>>(cls, su, hist1);
    k_findbucket<<<1, 1024, 0, stream>>>(hist1, nullptr, &stage[0], &stage[1]);
    k_hist2<<<N_ANCH / 256, 256, 0, stream>>>(su, stage, hist2);
    k_findbucket<<<1, 1024, 0, stream>>>(hist2, &stage[1], &stage[2], &stage[3]);
    k_collect<<<N_ANCH / 256, 256, 0, stream>>>(su, stage, cand, &stage[4]);
    k_sortgather<<<1, 1024, 0, stream>>>(cand, stage, boxes, topk, boxesk, areak);
    k_iou<<<dim3(64, 64), 32, 0, stream>>>(boxesk, areak, overH);
    k_scan<<<1, 32, 0, stream>>>(overH, topk, kept, validf);
    k_out<<<(MAX_BOXES * DET_DIM + 255) / 256, 256, 0, stream>>>(det, kept, validf, out);
}